// mTAN_Enc_46952582479932
// MI455X (gfx1250) — compile-verified
//
#include <hip/hip_runtime.h>
#include <math.h>

// Problem constants (from reference): B=8, L=512, K=16, E=32, H=1, NH=64, W_OFF=1
constexpr int CB  = 8;    // batch
constexpr int CL  = 512;  // sequence length
constexpr int CK  = 16;   // number of classes (marks)
constexpr int CE  = 32;   // embedding dim (== DK since H=1)
constexpr int CNH = 64;   // GRU hidden

typedef float v2f __attribute__((ext_vector_type(2)));
typedef float v8f __attribute__((ext_vector_type(8)));

// ---------------------------------------------------------------------------
// Kernel 1: sinusoidal time embedding + q/k projections.
// One 32-thread wave per (b,l); kemb staged in LDS; each lane produces one
// q component and one k component (dot over E=32).
// ---------------------------------------------------------------------------
__global__ __launch_bounds__(32)
void k1_embed_qk(const float* __restrict__ ts,
                 const float* __restrict__ Wl, const float* __restrict__ bl,
                 const float* __restrict__ Wp, const float* __restrict__ bp,
                 const float* __restrict__ Wq, const float* __restrict__ bq,
                 const float* __restrict__ Wk, const float* __restrict__ bk,
                 float* __restrict__ qbuf, float* __restrict__ kbuf)
{
    __shared__ float kemb[CE];
    const int bl_idx = blockIdx.x;        // b*L + l
    const int t = threadIdx.x;            // 0..31
    const float tt = ts[bl_idx];
    if (t == 0) kemb[0] = tt * Wl[0] + bl[0];
    else        kemb[t] = sinf(tt * Wp[t - 1] + bp[t - 1]);
    __syncthreads();
    float qv = bq[t];
    float kv = bk[t];
#pragma unroll
    for (int e = 0; e < CE; ++e) {
        qv += kemb[e] * Wq[t * CE + e];
        kv += kemb[e] * Wk[t * CE + e];
    }
    qbuf[(size_t)bl_idx * CE + t] = qv;
    kbuf[(size_t)bl_idx * CE + t] = kv;
}

// ---------------------------------------------------------------------------
// Kernel 2: scores via V_WMMA_F32_16X16X4_F32 + per-class streaming softmax.
// One wave per (batch, 16-row q tile). Computes the 16x512 score strip in LDS
// with 256 fp32 WMMAs, then lanes 0..15 run an online softmax with 16
// per-class accumulators (semantics match the reference: keys with k>q+1 but
// matching class contribute logit 0; non-matching keys' -1e9 underflows to 0).
// ---------------------------------------------------------------------------
__global__ __launch_bounds__(32)
void k2_attn(const float* __restrict__ qbuf, const float* __restrict__ kbuf,
             const float* __restrict__ val,  const int* __restrict__ mark,
             const float* __restrict__ npm,  float* __restrict__ xbuf)
{
    __shared__ float slds[16 * CL];          // 32 KB score strip
    __shared__ int   lm[CL];                 // class id per key (-1 = padded)
    __shared__ float lv[CL];                 // val per key
    __shared__ float amax[16 * CK];
    __shared__ float aden[16 * CK];
    __shared__ float anum[16 * CK];

    const int b   = blockIdx.x >> 5;         // 32 q-tiles per batch
    const int qt  = blockIdx.x & 31;
    const int t   = threadIdx.x;
    const int row = t & 15;
    const int hi  = t >> 4;                  // 0: K pair (4c,4c+1), 1: (4c+2,4c+3)

    // stage mark / val / pad-mask for this batch
    for (int k = t; k < CL; k += 32) {
        const int   m = mark[b * CL + k];
        const float p = npm[b * CL + k];
        lm[k] = (p > 0.f) ? (m - 1) : -1;
        lv[k] = val[b * CL + k];
    }

    // A operand (16x4 fp32 tiles of the q strip) held in registers, reused
    // across all 32 k-tiles.  ISA layout: lanes 0-15 -> M=lane, VGPR0=K0,
    // VGPR1=K1; lanes 16-31 -> M=lane-16, VGPR0=K2, VGPR1=K3.
    const float* qrow = qbuf + ((size_t)(b * CL + qt * 16 + row)) * CE;
    v2f aq[8];
#pragma unroll
    for (int c = 0; c < 8; ++c)
        aq[c] = *(const v2f*)(qrow + 4 * c + 2 * hi);

    const float scale = 0.17677669529663687f;   // 1/sqrt(32)
    for (int kt = 0; kt < 32; ++kt) {
        const float* krow = kbuf + ((size_t)(b * CL + kt * 16 + row)) * CE;
        v8f acc = {};
#pragma unroll
        for (int c = 0; c < 8; ++c) {
            v2f bk = *(const v2f*)(krow + 4 * c + 2 * hi);  // B = k^T tile
            acc = __builtin_amdgcn_wmma_f32_16x16x4_f32(
                false, aq[c], false, bk, (short)0, acc, false, false);
        }
        // D layout: VGPR i, lane t -> M = i + 8*(t>>4), N = t&15
#pragma unroll
        for (int i = 0; i < 8; ++i) {
            const int m = i + 8 * hi;
            const int n = kt * 16 + row;
            slds[m * CL + n] = acc[i] * scale;
        }
    }

    if (t < 16) {
#pragma unroll
        for (int c = 0; c < CK; ++c) {
            amax[t * CK + c] = -3.0e38f;
            aden[t * CK + c] = 0.f;
            anum[t * CK + c] = 0.f;
        }
    }
    __syncthreads();

    if (t < 16) {
        const int q = qt * 16 + t;
        for (int k = 0; k < CL; ++k) {
            const int c = lm[k];
            if (c < 0) continue;
            const float logit = (k <= q + 1) ? slds[t * CL + k] : 0.f;
            const float vv = lv[k];
            const float mo = amax[t * CK + c];
            if (logit > mo) {
                const float e = expf(mo - logit);     // 0 on first hit
                aden[t * CK + c] = aden[t * CK + c] * e + 1.f;
                anum[t * CK + c] = anum[t * CK + c] * e + vv;
                amax[t * CK + c] = logit;
            } else {
                const float e = expf(logit - mo);
                aden[t * CK + c] += e;
                anum[t * CK + c] += e * vv;
            }
        }
#pragma unroll
        for (int c = 0; c < CK; ++c) {
            const float d = aden[t * CK + c];
            xbuf[((size_t)(b * CL + q)) * CK + c] = (d > 0.f) ? anum[t * CK + c] / d : 0.f;
        }
    }
}

// ---------------------------------------------------------------------------
// Kernel 3: att_out = x @ Wo^T + bo, then gi = att_out @ W_ih^T + b_ih,
// fused per (b,l).  64 threads; att_out staged in LDS, never materialized.
// ---------------------------------------------------------------------------
__global__ __launch_bounds__(64)
void k3_proj_gates(const float* __restrict__ xbuf,
                   const float* __restrict__ Wo, const float* __restrict__ bo,
                   const float* __restrict__ W_ih, const float* __restrict__ b_ih,
                   float* __restrict__ gibuf)
{
    __shared__ float xv[CK];
    __shared__ float att[CNH];
    const int bl_idx = blockIdx.x;
    const int t = threadIdx.x;
    if (t < CK) xv[t] = xbuf[(size_t)bl_idx * CK + t];
    __syncthreads();
    float a = bo[t];
#pragma unroll
    for (int c = 0; c < CK; ++c) a += xv[c] * Wo[t * CK + c];
    att[t] = a;
    __syncthreads();
    for (int g = t; g < 3 * CNH; g += CNH) {
        float gv = b_ih[g];
#pragma unroll
        for (int u = 0; u < CNH; ++u) gv += att[u] * W_ih[g * CNH + u];
        gibuf[(size_t)bl_idx * (3 * CNH) + g] = gv;
    }
}

// ---------------------------------------------------------------------------
// Kernel 4: sequential GRU.  One block per batch, 192 threads (one per gate
// row); each thread caches its W_hh row in VGPRs; h lives in LDS.
// ---------------------------------------------------------------------------
__global__ __launch_bounds__(192)
void k4_gru(const float* __restrict__ gibuf,
            const float* __restrict__ W_hh, const float* __restrict__ b_hh,
            float* __restrict__ out)
{
    __shared__ float h[CNH];
    __shared__ float ghl[3 * CNH];
    const int b = blockIdx.x;
    const int j = threadIdx.x;    // 0..191

    float w[CNH];
#pragma unroll
    for (int u = 0; u < CNH; ++u) w[u] = W_hh[j * CNH + u];
    const float bh = b_hh[j];
    if (j < CNH) h[j] = 0.f;
    __syncthreads();

    for (int step = 0; step < CL; ++step) {
        float gh = bh;
#pragma unroll
        for (int u = 0; u < CNH; ++u) gh += w[u] * h[u];
        ghl[j] = gh;
        __syncthreads();    // gh ready; also all reads of h done (WAR safe)
        if (j < CNH) {
            const float* gi = gibuf + ((size_t)(b * CL + step)) * (3 * CNH);
            const float r  = 1.f / (1.f + expf(-(gi[j] + ghl[j])));
            const float z  = 1.f / (1.f + expf(-(gi[CNH + j] + ghl[CNH + j])));
            const float n  = tanhf(gi[2 * CNH + j] + r * ghl[2 * CNH + j]);
            const float hn = (1.f - z) * n + z * h[j];
            h[j] = hn;
            out[((size_t)(b * CL + step)) * CNH + j] = hn;
        }
        __syncthreads();    // h updated before next step reads it
    }
}

// ---------------------------------------------------------------------------
extern "C" void kernel_launch(void* const* d_in, const int* in_sizes, int n_in,
                              void* d_out, int out_size, void* d_ws, size_t ws_size,
                              hipStream_t stream)
{
    const float* ts   = (const float*)d_in[0];   // time_steps (B,L)
    const float* val  = (const float*)d_in[1];   // val        (B,L)
    const int*   mark = (const int*)  d_in[2];   // mark       (B,L)
    const float* npm  = (const float*)d_in[3];   // non_pad_mask (B,L,1)
    const float* Wl   = (const float*)d_in[4];
    const float* bl_  = (const float*)d_in[5];
    const float* Wp   = (const float*)d_in[6];
    const float* bp   = (const float*)d_in[7];
    const float* Wq   = (const float*)d_in[8];
    const float* bq   = (const float*)d_in[9];
    const float* Wk   = (const float*)d_in[10];
    const float* bk   = (const float*)d_in[11];
    const float* Wo   = (const float*)d_in[12];
    const float* bo   = (const float*)d_in[13];
    const float* W_ih = (const float*)d_in[14];
    const float* W_hh = (const float*)d_in[15];
    const float* b_ih = (const float*)d_in[16];
    const float* b_hh = (const float*)d_in[17];

    // Workspace layout (floats): q (B,L,32) | k (B,L,32) | x (B,L,16) | gi (B,L,192)
    float* ws    = (float*)d_ws;
    float* qbuf  = ws;
    float* kbuf  = qbuf + (size_t)CB * CL * CE;
    float* xbuf  = kbuf + (size_t)CB * CL * CE;
    float* gibuf = xbuf + (size_t)CB * CL * CK;

    k1_embed_qk<<<CB * CL, 32, 0, stream>>>(ts, Wl, bl_, Wp, bp, Wq, bq, Wk, bk,
                                            qbuf, kbuf);
    k2_attn<<<CB * (CL / 16), 32, 0, stream>>>(qbuf, kbuf, val, mark, npm, xbuf);
    k3_proj_gates<<<CB * CL, CNH, 0, stream>>>(xbuf, Wo, bo, W_ih, b_ih, gibuf);
    k4_gru<<<CB, 3 * CNH, 0, stream>>>(gibuf, W_hh, b_hh, (float*)d_out);
}